// YoloLoss_24369644437632
// MI455X (gfx1250) — compile-verified
//
#include <hip/hip_runtime.h>
#include <hip/hip_bf16.h>

#define SGRID 14.0f

typedef __attribute__((ext_vector_type(2))) float v2f;
typedef __attribute__((ext_vector_type(8))) float v8f;

// Wave32 reduction of two independent scalars using V_WMMA_F32_16X16X4_F32.
// A layout (CDNA5 ISA 16x4 f32): lane m (0-15) holds A[m,0],A[m,1] in its 2
// VGPRs; lane 16+m holds A[m,2],A[m,3].  So matrix row m = [a_m, b_m, a_{m+16},
// b_{m+16}].  B (4x16) selector: col0 = rows {0,2}, col1 = rows {1,3}:
//   D[m,0] = a_m + a_{m+16},  D[m,1] = b_m + b_{m+16}.
// Column n lives in lanes with (lane&15)==n; row m in VGPR m (lanes 0-15) or
// m-8 (lanes 16-31).  Summing the 8 D VGPRs + a lane-16 xor-shuffle completes
// the 32-lane sum; lane0 ends with sum(a), lane1 with sum(b).
__device__ __forceinline__ void wave_reduce_pair(float a, float b,
                                                 float& suma, float& sumb) {
  unsigned lane = threadIdx.x & 31u;
  v2f A;  A[0] = a; A[1] = b;
  v2f Bs; Bs[0] = ((lane & 15u) == 0u) ? 1.0f : 0.0f;
          Bs[1] = ((lane & 15u) == 1u) ? 1.0f : 0.0f;
  v8f C = {};
  v8f D = __builtin_amdgcn_wmma_f32_16x16x4_f32(false, A, false, Bs,
                                                (short)0, C, false, false);
  float s = D[0] + D[1] + D[2] + D[3] + D[4] + D[5] + D[6] + D[7];
  s += __shfl_xor(s, 16, 32);
  suma = __shfl(s, 0, 32);
  sumb = __shfl(s, 1, 32);
}

// Block-wide (256 thr = 8 waves) deterministic reduction of 4 components.
// Results end up in out4[0..3]; caller must __syncthreads() before reading.
__device__ __forceinline__ void block_reduce4(float c0, float c1, float c2,
                                              float c3, float* lds /*32 f*/,
                                              float* out4) {
  float s0, s1, s2, s3;
  wave_reduce_pair(c0, c1, s0, s1);
  wave_reduce_pair(c2, c3, s2, s3);
  unsigned wave = threadIdx.x >> 5;
  unsigned lane = threadIdx.x & 31u;
  if (lane == 0u) {
    lds[wave * 4 + 0] = s0; lds[wave * 4 + 1] = s1;
    lds[wave * 4 + 2] = s2; lds[wave * 4 + 3] = s3;
  }
  __syncthreads();
  if (threadIdx.x < 4u) {
    float acc = 0.0f;
#pragma unroll
    for (int w = 0; w < 8; ++w) acc += lds[w * 4 + threadIdx.x];
    out4[threadIdx.x] = acc;
  }
}

// Kernel 1: per-cell loss components -> per-block partial sums in ws.
// ws[blk*4 + {0:cls, 1:noobj, 2:contain, 3:reg}]
__global__ __launch_bounds__(256) void yolo_partial(
    const float* __restrict__ pred, const float* __restrict__ tbox,
    const float* __restrict__ tcls, const unsigned char* __restrict__ objmap,
    float* __restrict__ ws, int ncells) {
  __shared__ float lds[32];
  int cell = blockIdx.x * 256 + (int)threadIdx.x;

  float cls_l = 0.0f, noobj_l = 0.0f, cont_l = 0.0f, reg_l = 0.0f;
  if (cell < ncells) {
    const float2* p2 = reinterpret_cast<const float2*>(pred) + (size_t)cell * 15;
    const float4* tc4 = reinterpret_cast<const float4*>(tcls) + (size_t)cell * 5;
    float4 tb = reinterpret_cast<const float4*>(tbox)[cell];
    float m = objmap[cell] ? 1.0f : 0.0f;

    // pred boxes: ch0-4 box0, ch5-9 box1
    float2 q0 = p2[0], q1 = p2[1], q2 = p2[2], q3 = p2[3], q4 = p2[4];
    float x0 = q0.x, y0 = q0.y, w0 = q1.x, h0 = q1.y, c0 = q2.x;
    float x1 = q2.y, y1 = q3.x, w1 = q3.y, h1 = q4.x, c1 = q4.y;

    // class loss: ch10..29 vs target_cls
    float clssum = 0.0f;
#pragma unroll
    for (int i = 0; i < 5; ++i) {
      float4 t = tc4[i];
      float2 pa = p2[5 + 2 * i];
      float2 pb = p2[6 + 2 * i];
      float d0 = pa.x - t.x, d1 = pa.y - t.y;
      float d2 = pb.x - t.z, d3 = pb.y - t.w;
      clssum += d0 * d0 + d1 * d1 + d2 * d2 + d3 * d3;
    }
    cls_l = m * clssum;

    noobj_l = 0.5f * (1.0f - m) * (c0 * c0 + c1 * c1);

    const float invS = 1.0f / SGRID;
    float txc = tb.x * invS, tyc = tb.y * invS;
    float tx1 = txc - 0.5f * tb.z, ty1 = tyc - 0.5f * tb.w;
    float tx2 = txc + 0.5f * tb.z, ty2 = tyc + 0.5f * tb.w;
    float tarea = (tx2 - tx1) * (ty2 - ty1);

    auto iou_of = [&](float x, float y, float w, float h) -> float {
      float px = x * invS, py = y * invS;
      float ax = px - 0.5f * w, ay = py - 0.5f * h;
      float bx = px + 0.5f * w, by = py + 0.5f * h;
      float lx = fmaxf(ax, tx1), ly = fmaxf(ay, ty1);
      float rx = fminf(bx, tx2), ry = fminf(by, ty2);
      float iw = fmaxf(rx - lx, 0.0f), ih = fmaxf(ry - ly, 0.0f);
      float inter = iw * ih;
      float a1 = (bx - ax) * (by - ay);
      return inter / (a1 + tarea - inter);
    };
    float iou0 = iou_of(x0, y0, w0, h0);
    float iou1 = iou_of(x1, y1, w1, h1);
    bool take1 = iou1 > iou0;  // argmax ties -> box 0, matches jnp.argmax
    float bx = take1 ? x1 : x0, by = take1 ? y1 : y0;
    float bw = take1 ? w1 : w0, bh = take1 ? h1 : h0;
    float bc = take1 ? c1 : c0;
    float biou = take1 ? iou1 : iou0;

    float dx = bx - tb.x, dy = by - tb.y;
    float dw = sqrtf(bw) - sqrtf(tb.z);
    float dh = sqrtf(bh) - sqrtf(tb.w);
    reg_l = 5.0f * m * (dx * dx + dy * dy + dw * dw + dh * dh);

    float dc = bc - biou;
    cont_l = m * dc * dc;
  }

  // All threads arrive here together (EXEC all-1s for the WMMAs).
  block_reduce4(cls_l, noobj_l, cont_l, reg_l, lds, ws + (size_t)blockIdx.x * 4);
}

// Kernel 2: reduce nparts block partials, write the 5 outputs.
__global__ __launch_bounds__(256) void yolo_final(const float* __restrict__ ws,
                                                  float* __restrict__ out,
                                                  int nparts, int n_batch) {
  __shared__ float lds[32];
  __shared__ float fin[4];
  float c0 = 0.0f, c1 = 0.0f, c2 = 0.0f, c3 = 0.0f;
  for (int i = (int)threadIdx.x; i < nparts; i += 256) {
    c0 += ws[i * 4 + 0];
    c1 += ws[i * 4 + 1];
    c2 += ws[i * 4 + 2];
    c3 += ws[i * 4 + 3];
  }
  block_reduce4(c0, c1, c2, c3, lds, fin);
  __syncthreads();
  if (threadIdx.x == 0) {
    float cls = fin[0], noobj = fin[1], cont = fin[2], reg = fin[3];
    float total = (cls + noobj + cont + reg) / (float)n_batch;
    out[0] = total;
    out[1] = reg;
    out[2] = cont;
    out[3] = noobj;
    out[4] = cls;
  }
}

extern "C" void kernel_launch(void* const* d_in, const int* in_sizes, int n_in,
                              void* d_out, int out_size, void* d_ws,
                              size_t ws_size, hipStream_t stream) {
  (void)n_in; (void)out_size; (void)ws_size;
  const float* pred = (const float*)d_in[0];
  const float* tbox = (const float*)d_in[1];
  const float* tcls = (const float*)d_in[2];
  const unsigned char* objmap = (const unsigned char*)d_in[3];

  int ncells = in_sizes[3];           // N * S * S (bool map, 1 elem per cell)
  int n_batch = ncells / (14 * 14);   // N
  int nblocks = (ncells + 255) / 256; // 1568 for N=2048

  float* ws = (float*)d_ws;
  yolo_partial<<<nblocks, 256, 0, stream>>>(pred, tbox, tcls, objmap, ws, ncells);
  yolo_final<<<1, 256, 0, stream>>>(ws, (float*)d_out, nblocks, n_batch);
}